// SpikingColorVision_67018669686950
// MI455X (gfx1250) — compile-verified
//
#include <hip/hip_runtime.h>
#include <stdint.h>

// Izhikevich RS spiking color-vision scan for MI455X (gfx1250).
//
// Shape of the problem: 800-elem reduction -> 4 drives; then a strictly
// sequential 1000-step Euler scan over 32 neurons (4 pops x 8 ch); then a
// mean over channels. ~0.5 MFLOP, ~131 KB traffic: pure latency problem.
//
// Design:
//  * single workgroup of 256 threads (8 wave32s)
//  * all waves issue GLOBAL_LOAD_ASYNC_TO_LDS_B128 to stage the entire
//    noise array (substeps*32 floats = 128 KB, fits the 320 KB WGP LDS)
//    while simultaneously doing the drives reduction (overlap for free)
//  * s_wait_asynccnt 0 + barrier, then wave 0 (32 lanes = 32 neurons)
//    runs the 1000-step dependency chain reading noise via ds_load_b32
//  * channel means + opponency via wave32 __shfl_xor, lane 0 stores 6 floats

#define BLOCK 256
#define NLANES 32

__global__ __launch_bounds__(BLOCK)
void SpikingColorVision_kernel(const float* __restrict__ retL,
                               const float* __restrict__ retR,
                               const float* __restrict__ noise,
                               float* __restrict__ out,
                               int substeps, int nL, int nR)
{
    extern __shared__ float smem[];
    // smem[0..3]  : counts (food, enemy, rock, consp)
    // smem[4..7]  : drives (uv, blue, green, red)
    // smem[32..]  : staged noise, substeps*32 floats (16B-aligned region)
    float* counts = smem;
    float* drives = smem + 4;
    float* nbuf   = smem + 32;

    const int tid = threadIdx.x;

    if (tid < 4) counts[tid] = 0.0f;

    // ---- Kick off async global->LDS staging of the whole noise array.
    // CDNA5 async-copy path: per-lane 16B transfers, tracked by ASYNCcnt.
    {
        const uint32_t total_bytes = (uint32_t)substeps * NLANES * 4u;
        const uint32_t lds_base = (uint32_t)(uintptr_t)nbuf;      // low 32 bits = LDS offset
        const uint64_t gbase    = (uint64_t)(uintptr_t)noise;
        for (uint32_t off = (uint32_t)tid * 16u; off < total_bytes; off += BLOCK * 16u) {
            uint32_t laddr = lds_base + off;
            uint64_t gaddr = gbase + off;
            asm volatile("global_load_async_to_lds_b128 %0, %1, off"
                         :: "v"(laddr), "v"(gaddr) : "memory");
        }
    }

    __syncthreads();  // counts zeroed & visible before atomics

    // ---- Drives reduction over the 800 retinal type values (overlaps the
    // in-flight async copies above).
    {
        float f = 0.f, e = 0.f, r = 0.f, c = 0.f;
        const int n = nL + nR;
        for (int i = tid; i < n; i += BLOCK) {
            float t = (i < nL) ? retL[i] : retR[i - nL];
            f += (t > 0.7f)                ? 1.0f : 0.0f;
            e += (fabsf(t - 0.5f)  < 0.1f) ? 1.0f : 0.0f;
            r += (fabsf(t - 0.75f) < 0.1f) ? 1.0f : 0.0f;
            c += (fabsf(t - 0.25f) < 0.1f) ? 1.0f : 0.0f;
        }
        #pragma unroll
        for (int m = 16; m >= 1; m >>= 1) {
            f += __shfl_xor(f, m, 32);
            e += __shfl_xor(e, m, 32);
            r += __shfl_xor(r, m, 32);
            c += __shfl_xor(c, m, 32);
        }
        if ((tid & 31) == 0) {            // one lane per wave32
            atomicAdd(&counts[0], f);
            atomicAdd(&counts[1], e);
            atomicAdd(&counts[2], r);
            atomicAdd(&counts[3], c);
        }
    }

    // Each wave drains its own async transfers; barrier publishes LDS to all.
    asm volatile("s_wait_asynccnt 0" ::: "memory");
    __syncthreads();

    if (tid == 0) {
        float f = counts[0], e = counts[1], r = counts[2], c = counts[3];
        float inv = 1.0f / (f + e + r + c + 1e-8f);
        drives[0] = (f * 0.1f + e * 0.05f + c * 0.8f + r * 0.2f) * inv;  // uv
        drives[1] = (f * 0.2f + e * 0.1f  + c * 0.5f + r * 0.3f) * inv;  // blue
        drives[2] = (f * 0.8f + e * 0.3f  + c * 0.3f + r * 0.3f) * inv;  // green
        drives[3] = (f * 0.4f + e * 0.7f  + c * 0.2f + r * 0.3f) * inv;  // red
    }
    __syncthreads();

    // ---- Sequential Izhikevich scan: wave 0, one lane per neuron.
    if (tid < NLANES) {
        const int lane = tid;
        const int pop  = lane >> 3;                 // 0=uv 1=blue 2=green 3=red
        const float Ibase = drives[pop] * 10.0f - 2.0f;   // drive*10 + i_tonic

        float v    = -65.0f;
        float u    = 0.2f * -65.0f;
        float rate = 0.0f;

        #pragma unroll 4
        for (int t = 0; t < substeps; ++t) {
            float eps = nbuf[t * NLANES + lane];          // ds_load_b32
            float I   = Ibase + eps * 0.3f;
            float vn  = v + (0.04f * v * v + 5.0f * v + 140.0f - u + I);
            float un  = u + 0.02f * (0.2f * v - u);       // uses OLD v
            bool  sp  = vn >= 30.0f;
            v    = sp ? -65.0f : vn;
            u    = sp ? (un + 8.0f) : un;
            rate = 0.95f * rate + (sp ? 0.05f : 0.0f);
        }

        // Mean over the 8 channels of each population (8-lane groups).
        float s = rate;
        s += __shfl_xor(s, 1, 32);
        s += __shfl_xor(s, 2, 32);
        s += __shfl_xor(s, 4, 32);
        float mean = s * 0.125f;

        float m0 = __shfl(mean,  0, 32);   // uv
        float m1 = __shfl(mean,  8, 32);   // blue
        float m2 = __shfl(mean, 16, 32);   // green
        float m3 = __shfl(mean, 24, 32);   // red

        if (lane == 0) {
            out[0] = m0;
            out[1] = m1;
            out[2] = m2;
            out[3] = m3;
            out[4] = m3 - m2;                                // rg opponency
            out[5] = 0.5f * (m0 + m1) - 0.5f * (m2 + m3);    // by opponency
        }
    }
}

extern "C" void kernel_launch(void* const* d_in, const int* in_sizes, int n_in,
                              void* d_out, int out_size, void* d_ws, size_t ws_size,
                              hipStream_t stream) {
    (void)n_in; (void)out_size; (void)d_ws; (void)ws_size;
    const float* retL  = (const float*)d_in[0];   // (400,) f32
    const float* retR  = (const float*)d_in[1];   // (400,) f32
    const float* noise = (const float*)d_in[2];   // (substeps, 4, 8) f32
    float* out = (float*)d_out;                   // 6 f32

    const int nL = in_sizes[0];
    const int nR = in_sizes[1];
    const int substeps = in_sizes[2] / NLANES;    // noise is (substeps,4,8)

    // dynamic LDS: 32 floats of header + substeps*32 floats of staged noise
    const size_t shmem = (size_t)(32 + (size_t)substeps * NLANES) * sizeof(float);

    SpikingColorVision_kernel<<<1, BLOCK, shmem, stream>>>(
        retL, retR, noise, out, substeps, nL, nR);
}